// JPEGCompressionPRISAttack_13013750907032
// MI455X (gfx1250) — compile-verified
//
#include <hip/hip_runtime.h>
#include <math.h>

typedef __attribute__((ext_vector_type(2))) float v2f;
typedef __attribute__((ext_vector_type(8))) float v8f;

__constant__ float YTAB[64] = {
  16,11,10,16,24,40,51,61,
  12,12,14,19,26,58,60,55,
  14,13,16,24,40,57,69,56,
  14,17,22,29,51,87,80,62,
  18,22,37,56,68,109,103,77,
  24,35,55,64,81,104,113,92,
  49,64,78,87,103,121,120,101,
  72,92,95,98,112,100,103,99
};
__constant__ float CTAB[64] = {
  17,18,24,47,99,99,99,99,
  18,21,26,66,99,99,99,99,
  24,26,56,99,99,99,99,99,
  47,66,99,99,99,99,99,99,
  99,99,99,99,99,99,99,99,
  99,99,99,99,99,99,99,99,
  99,99,99,99,99,99,99,99,
  99,99,99,99,99,99,99,99
};

// Block-diagonal 16x16 DCT basis: D[x][u] = cos((2x+1)u*pi/16) within each
// 8x8 diagonal block, zero elsewhere. Precomputed -> no runtime cosf.
#define Z8 0.f,0.f,0.f,0.f,0.f,0.f,0.f,0.f
#define CR0 1.0f, 0.980785280f, 0.923879533f, 0.831469612f, 0.707106781f, 0.555570233f, 0.382683432f, 0.195090322f
#define CR1 1.0f, 0.831469612f, 0.382683432f,-0.195090322f,-0.707106781f,-0.980785280f,-0.923879533f,-0.555570233f
#define CR2 1.0f, 0.555570233f,-0.382683432f,-0.980785280f,-0.707106781f, 0.195090322f, 0.923879533f, 0.831469612f
#define CR3 1.0f, 0.195090322f,-0.923879533f,-0.555570233f, 0.707106781f, 0.831469612f,-0.382683432f,-0.980785280f
#define CR4 1.0f,-0.195090322f,-0.923879533f, 0.555570233f, 0.707106781f,-0.831469612f,-0.382683432f, 0.980785280f
#define CR5 1.0f,-0.555570233f,-0.382683432f, 0.980785280f,-0.707106781f,-0.195090322f, 0.923879533f,-0.831469612f
#define CR6 1.0f,-0.831469612f, 0.382683432f, 0.195090322f,-0.707106781f, 0.980785280f,-0.923879533f, 0.555570233f
#define CR7 1.0f,-0.980785280f, 0.923879533f,-0.831469612f, 0.707106781f,-0.555570233f, 0.382683432f,-0.195090322f
__constant__ float DTAB[256] = {
  CR0, Z8,  CR1, Z8,  CR2, Z8,  CR3, Z8,
  CR4, Z8,  CR5, Z8,  CR6, Z8,  CR7, Z8,
  Z8, CR0,  Z8, CR1,  Z8, CR2,  Z8, CR3,
  Z8, CR4,  Z8, CR5,  Z8, CR6,  Z8, CR7
};

// CDNA5 split-counter LDS wait; also a compiler memory barrier for the
// wave-synchronous LDS staging (single wave, DS ops are in-order per wave).
__device__ __forceinline__ void wsync() {
  asm volatile("s_wait_dscnt 0" ::: "memory");
}

__device__ __forceinline__ v8f wmma_f32(v2f a, v2f b, v8f c) {
  // V_WMMA_F32_16X16X4_F32: D = A(16x4) * B(4x16) + C(16x16)
  return __builtin_amdgcn_wmma_f32_16x16x4_f32(
      /*neg_a=*/false, a, /*neg_b=*/false, b,
      /*c_mod=*/(short)0, c, /*reuse_a=*/false, /*reuse_b=*/false);
}

// Full JPEG quant round-trip on a 16x16 tile (four 8x8 blocks at once):
//   coef = s .* (D^T X D); g = s*q*diff_round(coef/q); recon = D G D^T
// X given in WMMA B-operand chunks; result returned in C/D layout.
__device__ __forceinline__ v8f jpeg_tile(const v2f X[4], const v2f P[4],
                                         const v2f Q[4], float* stage,
                                         const float* qt, int lc, int hi) {
  const v8f z = {0.f, 0.f, 0.f, 0.f, 0.f, 0.f, 0.f, 0.f};
  // T1 = D^T * X
  v8f t1 = z;
#pragma unroll
  for (int k = 0; k < 4; ++k) t1 = wmma_f32(P[k], X[k], t1);
  // stage T1 (C layout), reload as A-operand chunks
#pragma unroll
  for (int r = 0; r < 8; ++r) stage[(r + (hi ? 8 : 0)) * 16 + lc] = t1[r];
  wsync();
  v2f a1[4];
#pragma unroll
  for (int k = 0; k < 4; ++k) {
    int kk = 4 * k + (hi ? 2 : 0);
    a1[k][0] = stage[lc * 16 + kk];
    a1[k][1] = stage[lc * 16 + kk + 1];
  }
  wsync();
  // coef = T1 * D
  v8f coef = z;
#pragma unroll
  for (int k = 0; k < 4; ++k) coef = wmma_f32(a1[k], P[k], coef);
  // pointwise: scale, quantize (diff_round), dequantize, fold IDCT scale.
  // In C/D layout the matrix row urow = r + 8*hi, so (urow & 7) == r for BOTH
  // lane halves -> alpha_u select folds at compile time; col v8 = lc & 7.
  const float is2 = 0.70710678118654752440f;
  const int v8i = lc & 7;
  const float sv = 0.25f * (v8i == 0 ? is2 : 1.0f);
  v8f g;
#pragma unroll
  for (int r = 0; r < 8; ++r) {
    const float s = (r == 0) ? sv * is2 : sv;
    const float q = qt[r * 8 + v8i] * 0.4f;  // FACTOR for QUALITY=80
    const float rq = __builtin_amdgcn_rcpf(q);
    const float t = (s * coef[r]) * rq;
    const float rr = rintf(t);
    const float d = t - rr;
    g[r] = (s * q) * (rr + d * d * d);
  }
  // stage G, reload as B-operand chunks
#pragma unroll
  for (int r = 0; r < 8; ++r) stage[(r + (hi ? 8 : 0)) * 16 + lc] = g[r];
  wsync();
  v2f bg[4];
#pragma unroll
  for (int k = 0; k < 4; ++k) {
    int kk = 4 * k + (hi ? 2 : 0);
    bg[k][0] = stage[kk * 16 + lc];
    bg[k][1] = stage[(kk + 1) * 16 + lc];
  }
  wsync();
  // T2 = D * G
  v8f t2 = z;
#pragma unroll
  for (int k = 0; k < 4; ++k) t2 = wmma_f32(Q[k], bg[k], t2);
  // stage T2, reload as A-operand chunks
#pragma unroll
  for (int r = 0; r < 8; ++r) stage[(r + (hi ? 8 : 0)) * 16 + lc] = t2[r];
  wsync();
  v2f a2[4];
#pragma unroll
  for (int k = 0; k < 4; ++k) {
    int kk = 4 * k + (hi ? 2 : 0);
    a2[k][0] = stage[lc * 16 + kk];
    a2[k][1] = stage[lc * 16 + kk + 1];
  }
  wsync();
  // recon = T2 * D^T
  v8f rec = z;
#pragma unroll
  for (int k = 0; k < 4; ++k) rec = wmma_f32(a2[k], Q[k], rec);
  return rec;
}

__global__ __launch_bounds__(256) void jpeg_prisattack_kernel(
    const float* __restrict__ in, float* __restrict__ out) {
  __shared__ float sY[8][256];
  __shared__ float sCb[8][256];
  __shared__ float sCr[8][256];
  __shared__ float sStage[8][256];

  const int tid = threadIdx.x;
  const int w = tid >> 5;
  const int l = tid & 31;
  const int lc = l & 15;
  const int hi = l >> 4;

  const int region = blockIdx.x * 8 + w;  // 32768 regions total
  const int b = region >> 10;             // image (32x32 tiles per image)
  const int rem = region & 1023;
  const int r0 = (rem >> 5) * 16;
  const int c0 = (rem & 31) * 16;

  float* stage = sStage[w];
  float* yT = sY[w];
  float* cbT = sCb[w];
  float* crT = sCr[w];

  const long chstride = 512L * 512L;

  // ---- cooperative b128 load of RGB 16x16 region; RGB->YCbCr to LDS ----
  // (x255 folded into the conversion coefficients)
  {
    const int qrow = l >> 2;
    const int col4 = (l & 3) * 4;
#pragma unroll
    for (int i = 0; i < 2; ++i) {
      const int row = qrow + i * 8;
      const long base = (((long)b * 3) * 512 + (r0 + row)) * 512 + c0 + col4;
      const float4 rv = *(const float4*)(in + base);
      const float4 gv = *(const float4*)(in + base + chstride);
      const float4 bv = *(const float4*)(in + base + 2 * chstride);
      float rr[4] = {rv.x, rv.y, rv.z, rv.w};
      float gg[4] = {gv.x, gv.y, gv.z, gv.w};
      float bb[4] = {bv.x, bv.y, bv.z, bv.w};
      float y4[4], cb4[4], cr4[4];
#pragma unroll
      for (int j = 0; j < 4; ++j) {
        const float R = rr[j], G = gg[j], B = bb[j];
        y4[j]  =  76.245f    * R + 149.685f   * G +  29.07f    * B;
        cb4[j] = -43.02768f  * R -  84.47232f * G + 127.5f     * B;  // centered
        cr4[j] =  127.5f     * R - 106.76544f * G -  20.73456f * B;  // centered
      }
      *(float4*)&yT[row * 16 + col4]  = make_float4(y4[0], y4[1], y4[2], y4[3]);
      *(float4*)&cbT[row * 16 + col4] = make_float4(cb4[0], cb4[1], cb4[2], cb4[3]);
      *(float4*)&crT[row * 16 + col4] = make_float4(cr4[0], cr4[1], cr4[2], cr4[3]);
    }
  }
  wsync();

  // ---- DCT operand chunks from constant table ----
  // A-layout of D^T == B-layout of D (P); A-layout of D == B-layout of D^T (Q)
  v2f P[4], Q[4];
#pragma unroll
  for (int k = 0; k < 4; ++k) {
    int kk = 4 * k + (hi ? 2 : 0);
    P[k][0] = DTAB[kk * 16 + lc];
    P[k][1] = DTAB[(kk + 1) * 16 + lc];
    Q[k][0] = DTAB[lc * 16 + kk];
    Q[k][1] = DTAB[lc * 16 + kk + 1];
  }

  // ---- X operand chunks: Y tile (minus 128) and block-diag(Cb_s, Cr_s) ----
  v2f XY[4], XC[4];
#pragma unroll
  for (int k = 0; k < 4; ++k) {
    int kk = 4 * k + (hi ? 2 : 0);
    XY[k][0] = yT[kk * 16 + lc] - 128.f;
    XY[k][1] = yT[(kk + 1) * 16 + lc] - 128.f;
#pragma unroll
    for (int j = 0; j < 2; ++j) {
      int rowc = kk + j;
      float v = 0.f;
      if (rowc < 8 && lc < 8) {  // Cb quadrant: 2x2 mean downsample
        int i2 = rowc * 2, j2 = lc * 2;
        v = 0.25f * (cbT[i2 * 16 + j2] + cbT[i2 * 16 + j2 + 1] +
                     cbT[(i2 + 1) * 16 + j2] + cbT[(i2 + 1) * 16 + j2 + 1]);
      } else if (rowc >= 8 && lc >= 8) {  // Cr quadrant
        int i2 = (rowc - 8) * 2, j2 = (lc - 8) * 2;
        v = 0.25f * (crT[i2 * 16 + j2] + crT[i2 * 16 + j2 + 1] +
                     crT[(i2 + 1) * 16 + j2] + crT[(i2 + 1) * 16 + j2 + 1]);
      }
      XC[k][j] = v;
    }
  }
  wsync();

  // ---- two WMMA pipelines: luma (4 blocks) and chroma (Cb+Cr blocks) ----
  v8f yRec = jpeg_tile(XY, P, Q, stage, YTAB, lc, hi);
  v8f cRec = jpeg_tile(XC, P, Q, stage, CTAB, lc, hi);

  // publish chroma recon (still centered) to LDS for upsample lookup
#pragma unroll
  for (int r = 0; r < 8; ++r) stage[(r + (hi ? 8 : 0)) * 16 + lc] = cRec[r];
  wsync();

  // ---- YCbCr -> RGB, clip, /255; stage planes for vectorized stores ----
#pragma unroll
  for (int r = 0; r < 8; ++r) {
    const int x = r + (hi ? 8 : 0);
    const int c = lc;
    const float yv = yRec[r] + 128.f;
    const float cbv = stage[(x >> 1) * 16 + (c >> 1)];
    const float crv = stage[((x >> 1) + 8) * 16 + ((c >> 1) + 8)];
    float R = yv + 1.402f * crv;
    float G = yv - 0.344136f * cbv - 0.714136f * crv;
    float B = yv + 1.772f * cbv;
    R = fminf(fmaxf(R, 0.f), 255.f) * (1.0f / 255.0f);
    G = fminf(fmaxf(G, 0.f), 255.f) * (1.0f / 255.0f);
    B = fminf(fmaxf(B, 0.f), 255.f) * (1.0f / 255.0f);
    yT[x * 16 + c] = R;
    cbT[x * 16 + c] = G;
    crT[x * 16 + c] = B;
  }
  wsync();

  // ---- cooperative b128 stores ----
  {
    const int qrow = l >> 2;
    const int col4 = (l & 3) * 4;
#pragma unroll
    for (int i = 0; i < 2; ++i) {
      const int row = qrow + i * 8;
      const long base = (((long)b * 3) * 512 + (r0 + row)) * 512 + c0 + col4;
      *(float4*)(out + base) = *(const float4*)&yT[row * 16 + col4];
      *(float4*)(out + base + chstride) = *(const float4*)&cbT[row * 16 + col4];
      *(float4*)(out + base + 2 * chstride) = *(const float4*)&crT[row * 16 + col4];
    }
  }
}

extern "C" void kernel_launch(void* const* d_in, const int* in_sizes, int n_in,
                              void* d_out, int out_size, void* d_ws,
                              size_t ws_size, hipStream_t stream) {
  (void)in_sizes; (void)n_in; (void)d_ws; (void)ws_size; (void)out_size;
  const float* in = (const float*)d_in[0];
  float* out = (float*)d_out;
  // 32 images * (512/16)^2 tiles = 32768 regions; 8 waves per 256-thread block
  jpeg_prisattack_kernel<<<4096, 256, 0, stream>>>(in, out);
}